// Attention_88278757802132
// MI455X (gfx1250) — compile-verified
//
#include <hip/hip_runtime.h>
#include <cstdint>

// Shapes fixed by the reference harness.
#define HEADS 8
#define DHEAD 64
#define INNER 512   // HEADS*DHEAD
#define NE    256
#define EE    128
#define NN    384
#define BB    2
#define BN    (BB*NN)
#define E_STRIDE 132   // 132 % 64 banks = 4 -> conflict-free WMMA A-row gathers

typedef __attribute__((ext_vector_type(2))) float v2f;
typedef __attribute__((ext_vector_type(8))) float v8f;

// CDNA5 async global->LDS path (ASYNCcnt-tracked); fall back to float4 copies
// if this toolchain does not declare the builtin. Probe round 2 revealed the
// parameter type: pointer to 16-byte int vector in the device (AS1) space.
#if __has_builtin(__builtin_amdgcn_global_load_async_to_lds_b128)
#define USE_ASYNC_LDS 1
typedef int gv4i __attribute__((vector_size(16)));
typedef gv4i __attribute__((address_space(1)))* gv4i_g;   // printed as "__device__ *"
typedef gv4i __attribute__((address_space(3)))* gv4i_s;   // printed as "__shared__ *"
__device__ __forceinline__ void wait_async0() {
#if __has_builtin(__builtin_amdgcn_s_wait_asynccnt)
  __builtin_amdgcn_s_wait_asynccnt(0);
#else
  asm volatile("s_wait_asynccnt 0x0" ::: "memory");
#endif
}
#else
#define USE_ASYNC_LDS 0
#endif

// ---------------------------------------------------------------------------
// Kernel A: per-row projections. Q,K,V = nodes@{Wq,Wkv}; U[h,c] = We_h^T q_h
// (folds q.e_ij into an edges.u dot so e is never materialized).
// ---------------------------------------------------------------------------
__global__ __launch_bounds__(256) void proj_qkvu(
    const float* __restrict__ nodes, const float* __restrict__ Wq, const float* __restrict__ bq,
    const float* __restrict__ Wkv, const float* __restrict__ bkv, const float* __restrict__ We,
    float* __restrict__ Q, float* __restrict__ K, float* __restrict__ V, float* __restrict__ U)
{
  __shared__ float nl[NE];
  __shared__ float qs[INNER];
  const int t = threadIdx.x;
  const int row = blockIdx.x;           // row = b*NN + i
  if (t < NE) nl[t] = nodes[(size_t)row*NE + t];
  __syncthreads();
  #pragma unroll
  for (int i = 0; i < 2; ++i) {
    const int o = t + 256*i;
    float acc = bq[o];
    #pragma unroll 4
    for (int c = 0; c < NE; ++c) acc += nl[c] * Wq[(size_t)c*INNER + o];
    qs[o] = acc;
    Q[(size_t)row*INNER + o] = acc;
  }
  #pragma unroll
  for (int i = 0; i < 4; ++i) {
    const int o = t + 256*i;
    float acc = bkv[o];
    #pragma unroll 4
    for (int c = 0; c < NE; ++c) acc += nl[c] * Wkv[(size_t)c*2*INNER + o];
    if (o < INNER) K[(size_t)row*INNER + o] = acc;
    else           V[(size_t)row*INNER + (o - INNER)] = acc;
  }
  __syncthreads();
  #pragma unroll
  for (int i = 0; i < 4; ++i) {
    const int o = t + 256*i;            // o = h*128 + c
    const int h = o >> 7, c = o & 127;
    const float* qh = &qs[h*DHEAD];
    const float* wr = &We[(size_t)c*INNER + h*DHEAD];
    float acc = 0.f;
    #pragma unroll
    for (int d = 0; d < DHEAD; d += 4) {
      float4 a = *(const float4*)(qh + d);
      float4 w = *(const float4*)(wr + d);
      acc += a.x*w.x + a.y*w.y + a.z*w.z + a.w*w.w;
    }
    U[(size_t)row*(HEADS*EE) + o] = acc;
  }
}

// ---------------------------------------------------------------------------
// Kernel A2: sim_qk = Q @ K^T per (b,h)  — WMMA f32 batched GEMM.
// Hoisted out of the fused kernel: K_b is now read ~24x total instead of 768x.
// ---------------------------------------------------------------------------
__global__ __launch_bounds__(256) void qk_gemm(
    const float* __restrict__ Qg, const float* __restrict__ Kg, float* __restrict__ S)
{
  __shared__ float a_lds[16*68];        // 68 % 64 banks = 4 -> no conflicts
  const int t = threadIdx.x, lane = t & 31, wave = t >> 5;
  const int l16 = lane & 15, kHi = (lane >> 4) << 1;
  const int bh = blockIdx.y, b = bh >> 3, h = bh & 7;
  const int i0 = blockIdx.x * 16;
  #pragma unroll
  for (int r = 0; r < 4; ++r) {
    const int o = t + 256*r;            // 16x64 A tile
    const int m = o >> 6, d = o & 63;
    a_lds[m*68 + d] = Qg[((size_t)(b*NN + i0 + m))*INNER + h*DHEAD + d];
  }
  __syncthreads();
  #pragma unroll
  for (int ts = 0; ts < 3; ++ts) {
    const int j0 = (wave + 8*ts) * 16;
    const int j = j0 + l16;
    const float* brow = Kg + ((size_t)(b*NN + j))*INNER + h*DHEAD;  // B col n=j
    const float* arow = a_lds + l16*68;                              // A row m=i
    v8f acc = {};
    #pragma unroll
    for (int k0 = 0; k0 < DHEAD; k0 += 4) {
      v2f a  = *(const v2f*)(arow + k0 + kHi);
      v2f bb = *(const v2f*)(brow + k0 + kHi);
      acc = __builtin_amdgcn_wmma_f32_16x16x4_f32(false, a, false, bb,
                                                  (short)0, acc, false, false);
    }
    const int mBase = i0 + ((lane >> 4) << 3);
    float* srow = S + ((size_t)bh*NN)*NN;
    #pragma unroll
    for (int r = 0; r < 8; ++r)
      srow[(size_t)(mBase + r)*NN + j] = acc[r];
  }
}

// ---------------------------------------------------------------------------
// Kernel B: fused edge-attention per (b,i). edges[b,i,:,:] is async-DMA'd ONCE
// into LDS (384x128 f32, padded stride 132) and consumed by two WMMA GEMMs:
//   sim += E @ U^T   and   w = attn @ E.
// Reads the qk row from S and overwrites it in place with attn for av_gemm.
// ---------------------------------------------------------------------------
__global__ __launch_bounds__(256) void edge_attn(
    const float* __restrict__ edges, const unsigned char* __restrict__ mask,
    float* __restrict__ S, const float* __restrict__ Ug,
    const float* __restrict__ We, const float* __restrict__ be,
    float* __restrict__ O)
{
  extern __shared__ float sm[];
  float* Elds = sm;                       // [384][132]
  float* simL = Elds + NN*E_STRIDE;       // [16][384]  rows 8..15 zero (WMMA A pad)
  float* wl   = simL + 16*NN;             // [8][128]
  float* Ul   = wl + HEADS*EE;            // [16][128]  rows 8..15 zero (WMMA B pad)

  const int t    = threadIdx.x;
  const int lane = t & 31, wave = t >> 5;
  const int l16  = lane & 15;
  const int kHi  = (lane >> 4) << 1;      // per-lane K sub-slot of 16x16x4 f32 operands
  const int bi   = blockIdx.x;
  const int bb   = bi / NN, qi = bi - bb*NN;

  // ---- phase 0: async E fill; stage qk row + U; zero WMMA padding --------
  const float* Eg = edges + (size_t)bi * NN * EE;
#if USE_ASYNC_LDS
  #pragma unroll 4
  for (int it = 0; it < 48; ++it) {       // 48 wave-instrs -> ASYNCcnt <= 48 < 63
    const int v = t + 256*it;
    const int j = v >> 5, c4 = v & 31;
    __builtin_amdgcn_global_load_async_to_lds_b128(
        (gv4i_g)(Eg + (size_t)j*EE + c4*4),
        (gv4i_s)(&Elds[j*E_STRIDE + c4*4]), 0, 0);
  }
#else
  #pragma unroll 4
  for (int it = 0; it < 48; ++it) {
    const int v = t + 256*it;
    const int j = v >> 5, c4 = v & 31;
    float4 x = *(const float4*)(Eg + (size_t)j*EE + c4*4);
    *(float4*)(&Elds[j*E_STRIDE + c4*4]) = x;
  }
#endif
  for (int idx = t; idx < 8*NN; idx += 256) simL[8*NN + idx] = 0.f;
  for (int idx = t; idx < 8*EE; idx += 256) Ul[8*EE + idx] = 0.f;
  #pragma unroll
  for (int i = 0; i < 4; ++i)
    Ul[t + 256*i] = Ug[(size_t)bi*(HEADS*EE) + t + 256*i];
  #pragma unroll
  for (int it = 0; it < 12; ++it) {       // qk row: s enumerates h*NN+j directly
    const int s = t + 256*it;
    const int h = s / NN, j = s - h*NN;
    simL[s] = S[((size_t)(bb*HEADS + h)*NN + qi)*NN + j];
  }
  if (t < EE) __builtin_prefetch(We + (size_t)t*INNER, 0, 1);  // global_prefetch_b8
#if USE_ASYNC_LDS
  wait_async0();                          // own waves' ASYNCcnt -> 0
#endif
  __syncthreads();                        // all waves' E + qk visible

  // ---- phase 1: sim += E @ U^T  (WMMA f32, D[m=j][n=h]) ------------------
  #pragma unroll
  for (int tile = 0; tile < 3; ++tile) {
    const int j0 = (wave + 8*tile) * 16;
    const float* Ea = Elds + (size_t)(j0 + l16)*E_STRIDE;   // A row m=j
    const float* Ub = Ul + l16*EE;                          // B col n=h (padded)
    v8f acc = {};
    #pragma unroll
    for (int k0 = 0; k0 < EE; k0 += 4) {
      v2f a = *(const v2f*)(Ea + k0 + kHi);
      v2f b = *(const v2f*)(Ub + k0 + kHi);
      acc = __builtin_amdgcn_wmma_f32_16x16x4_f32(false, a, false, b,
                                                  (short)0, acc, false, false);
    }
    if (l16 < 8) {                        // n>=8 is head padding
      const int mBase = j0 + ((lane >> 4) << 3);
      #pragma unroll
      for (int r = 0; r < 8; ++r) simL[l16*NN + mBase + r] += acc[r];
    }
  }
  __syncthreads();

  // ---- phase 2: softmax over j; write attn to LDS and back to S ----------
  {
    const int h = wave;
    const float scale = 0.125f;           // d^-0.5
    const unsigned char mi = mask[bb*NN + qi];
    float* srow = S + ((size_t)(bb*HEADS + h)*NN + qi)*NN;
    float vals[12];
    float mx = -3.0e38f;
    #pragma unroll
    for (int it = 0; it < 12; ++it) {
      const int j = lane + 32*it;
      const unsigned char mj = mask[bb*NN + j];
      float s = (mi && mj) ? simL[h*NN + j]*scale : -3.0e38f;
      vals[it] = s; mx = fmaxf(mx, s);
    }
    #pragma unroll
    for (int s = 16; s >= 1; s >>= 1) mx = fmaxf(mx, __shfl_xor(mx, s, 32));
    float sum = 0.f;
    #pragma unroll
    for (int it = 0; it < 12; ++it) { float e = __expf(vals[it] - mx); vals[it] = e; sum += e; }
    #pragma unroll
    for (int s = 16; s >= 1; s >>= 1) sum += __shfl_xor(sum, s, 32);
    const float inv = 1.0f / sum;
    #pragma unroll
    for (int it = 0; it < 12; ++it) {
      const int j = lane + 32*it;
      const float a = vals[it]*inv;
      simL[h*NN + j] = a;
      srow[j] = a;                        // attn for av_gemm
    }
  }
  __syncthreads();

  // ---- phase 3: w = attn @ E  (WMMA f32, D[m=h][n=c]) --------------------
  {
    const int c = wave*16 + l16;          // wave owns one 16-col N-tile
    const float* Ar = simL + l16*NN;      // A row m=h (rows 8..15 zero pad)
    v8f acc = {};
    #pragma unroll 8
    for (int k0 = 0; k0 < NN; k0 += 4) {
      const int k = k0 + kHi;
      v2f a = *(const v2f*)(Ar + k);
      v2f b; b.x = Elds[(size_t)k*E_STRIDE + c];
             b.y = Elds[(size_t)(k+1)*E_STRIDE + c];
      acc = __builtin_amdgcn_wmma_f32_16x16x4_f32(false, a, false, b,
                                                  (short)0, acc, false, false);
    }
    if ((lane >> 4) == 0) {               // upper-lane rows are head padding
      #pragma unroll
      for (int r = 0; r < 8; ++r) wl[r*EE + c] = acc[r];
    }
  }
  __syncthreads();

  // ---- phase 4: O = w @ We_h + be  (attn@V added later by av_gemm) -------
  {
    const int h = wave, d0 = lane*2;
    float a0 = 0.f, a1 = 0.f;
    const float* webase = We + h*DHEAD + d0;
    #pragma unroll 4
    for (int c = 0; c < EE; ++c) {
      const float wv = wl[h*EE + c];
      float2 w2 = *(const float2*)(webase + (size_t)c*INNER);
      a0 += wv*w2.x; a1 += wv*w2.y;
    }
    float2 b2 = *(const float2*)(be + h*DHEAD + d0);  // sum(attn)=1 -> be passes through
    O[(size_t)bi*INNER + h*DHEAD + d0]     = a0 + b2.x;
    O[(size_t)bi*INNER + h*DHEAD + d0 + 1] = a1 + b2.y;
  }
}

// ---------------------------------------------------------------------------
// Kernel C2: O += attn @ V per (b,h) — WMMA f32 batched GEMM (V read ~19 MB
// total instead of 576 MB inside the fused kernel).
// ---------------------------------------------------------------------------
__global__ __launch_bounds__(256) void av_gemm(
    const float* __restrict__ S, const float* __restrict__ Vg, float* __restrict__ O)
{
  __shared__ float a_lds[32*388];         // 388 % 64 banks = 4 -> no conflicts
  const int t = threadIdx.x, lane = t & 31, wave = t >> 5;
  const int l16 = lane & 15, kHi = (lane >> 4) << 1;
  const int bh = blockIdx.y, b = bh >> 3, h = bh & 7;
  const int i0 = blockIdx.x * 32;
  #pragma unroll
  for (int r = 0; r < 12; ++r) {          // stage attn tile [32 x 384]
    const int v = t + 256*r;
    const int m = v / 96, c4 = v - m*96;
    float4 x = *(const float4*)(S + ((size_t)(bh*NN + i0 + m))*NN + c4*4);
    *(float4*)(&a_lds[m*388 + c4*4]) = x;
  }
  __syncthreads();
  const int m0 = (wave >> 2) * 16, d0 = (wave & 3) * 16;
  const float* arow = a_lds + (m0 + l16)*388;
  v8f acc = {};
  #pragma unroll 8
  for (int k0 = 0; k0 < NN; k0 += 4) {
    const int k = k0 + kHi;
    v2f a = *(const v2f*)(arow + k);
    v2f bb;
    bb.x = Vg[((size_t)(b*NN + k))*INNER + h*DHEAD + d0 + l16];
    bb.y = Vg[((size_t)(b*NN + k + 1))*INNER + h*DHEAD + d0 + l16];
    acc = __builtin_amdgcn_wmma_f32_16x16x4_f32(false, a, false, bb,
                                                (short)0, acc, false, false);
  }
  const int mBase = i0 + m0 + ((lane >> 4) << 3);
  #pragma unroll
  for (int r = 0; r < 8; ++r) {
    const size_t idx = ((size_t)(b*NN + mBase + r))*INNER + h*DHEAD + d0 + l16;
    O[idx] += acc[r];                     // each element owned by one lane
  }
}

// ---------------------------------------------------------------------------
// Kernel D: final projection  out = O @ Wo + bo   (Wo is L2-resident)
// ---------------------------------------------------------------------------
__global__ __launch_bounds__(256) void out_proj(
    const float* __restrict__ O, const float* __restrict__ Wo,
    const float* __restrict__ bo, float* __restrict__ out)
{
  __shared__ float ol[INNER];
  const int t = threadIdx.x;
  const int row = blockIdx.x;
  ol[t]       = O[(size_t)row*INNER + t];
  ol[t + 256] = O[(size_t)row*INNER + 256 + t];
  __syncthreads();
  float acc = bo[t];
  #pragma unroll 4
  for (int p = 0; p < INNER; ++p) acc += ol[p] * Wo[(size_t)p*NE + t];
  out[(size_t)row*NE + t] = acc;
}

// ---------------------------------------------------------------------------
extern "C" void kernel_launch(void* const* d_in, const int* in_sizes, int n_in,
                              void* d_out, int out_size, void* d_ws, size_t ws_size,
                              hipStream_t stream)
{
  const float* nodes = (const float*)d_in[0];
  const float* edges = (const float*)d_in[1];
  const unsigned char* mask = (const unsigned char*)d_in[2];   // jnp bool -> 1 byte
  const float* Wq  = (const float*)d_in[3];
  const float* bq  = (const float*)d_in[4];
  const float* Wkv = (const float*)d_in[5];
  const float* bkv = (const float*)d_in[6];
  const float* We  = (const float*)d_in[7];
  const float* be  = (const float*)d_in[8];
  const float* Wo  = (const float*)d_in[9];
  const float* bo  = (const float*)d_in[10];

  float* ws = (float*)d_ws;                    // ~19 MB fp32 scratch
  float* Q = ws;
  float* K = Q + (size_t)BN*INNER;
  float* V = K + (size_t)BN*INNER;
  float* U = V + (size_t)BN*INNER;
  float* S = U + (size_t)BN*HEADS*EE;          // [16][384][384]: qk -> attn in place
  float* O = S + (size_t)BB*HEADS*NN*NN;

  const size_t smem = (size_t)(NN*E_STRIDE + 16*NN + HEADS*EE + 16*EE)
                      * sizeof(float);         // ~234 KB of the 320 KB WGP LDS
  (void)hipFuncSetAttribute(reinterpret_cast<const void*>(edge_attn),
                            hipFuncAttributeMaxDynamicSharedMemorySize, (int)smem);

  proj_qkvu<<<BN, 256, 0, stream>>>(nodes, Wq, bq, Wkv, bkv, We, Q, K, V, U);
  qk_gemm<<<dim3(24, 16), 256, 0, stream>>>(Q, K, S);
  edge_attn<<<BN, 256, smem, stream>>>(edges, mask, S, U, We, be, O);
  av_gemm<<<dim3(12, 16), 256, 0, stream>>>(S, V, O);
  out_proj<<<BN, 256, 0, stream>>>(O, Wo, bo, (float*)d_out);
}